// DINOCLIPEncoderMatching_75754633167081
// MI455X (gfx1250) — compile-verified
//
#include <hip/hip_runtime.h>

// Problem constants (from reference)
#define BB   2
#define CC   16
#define HH   98
#define WW   168
#define DD   96
#define HWN  (HH * WW)          // 16464
#define DCHUNK 32               // depths per thread; gridDim.y = DD/DCHUNK = 3
#define TPB  256                // threads (pixels) per block

static constexpr float MIN_DEPTH = 0.1f;
static constexpr float MAX_DEPTH = 20.0f;
static constexpr float DSTEP     = (MAX_DEPTH - MIN_DEPTH) / (float)(DD - 1);
static constexpr float EPS       = 1e-7f;

// gfx1250 async global->LDS copy path (probe via __has_builtin; host pass and
// older toolchains fall back to direct VGPR loads).
#if defined(__has_builtin)
#if __has_builtin(__builtin_amdgcn_global_load_async_to_lds_b128)
#define HAVE_ASYNC_LDS 1
#endif
#endif
#ifndef HAVE_ASYNC_LDS
#define HAVE_ASYNC_LDS 0
#endif

#if HAVE_ASYNC_LDS
typedef int v4i __attribute__((ext_vector_type(4)));
typedef __attribute__((address_space(1))) v4i GV4;     // global v4i
typedef __attribute__((address_space(3))) v4i LV4;     // LDS v4i

__device__ __forceinline__ void async_copy_b128(const void* gsrc, void* ldst) {
    GV4* g = (GV4*)(unsigned long long)gsrc;
    LV4* l = (LV4*)(unsigned int)(unsigned long long)ldst;  // low 32b = LDS byte offset
    __builtin_amdgcn_global_load_async_to_lds_b128(g, l, 0, 0);
}

__device__ __forceinline__ void wait_async_zero() {
#if __has_builtin(__builtin_amdgcn_s_wait_asynccnt)
    __builtin_amdgcn_s_wait_asynccnt(0);
#else
    asm volatile("s_wait_asynccnt 0x0" ::: "memory");
#endif
}
#endif

__device__ __forceinline__ int iclamp(int v, int lo, int hi) {
    return v < lo ? lo : (v > hi ? hi : v);
}

// Accumulate |bilinear(4 taps) - cur| for 4 channels packed in float4s.
__device__ __forceinline__ void acc_tap4(const float4 p00, const float4 p10,
                                         const float4 p01, const float4 p11,
                                         const float c0, const float c1,
                                         const float c2, const float c3,
                                         const float w00, const float w10,
                                         const float w01, const float w11,
                                         float& sum) {
    sum += fabsf(fmaf(w00, p00.x, fmaf(w10, p10.x, fmaf(w01, p01.x, w11 * p11.x))) - c0);
    sum += fabsf(fmaf(w00, p00.y, fmaf(w10, p10.y, fmaf(w01, p01.y, w11 * p11.y))) - c1);
    sum += fabsf(fmaf(w00, p00.z, fmaf(w10, p10.z, fmaf(w01, p01.z, w11 * p11.z))) - c2);
    sum += fabsf(fmaf(w00, p00.w, fmaf(w10, p10.w, fmaf(w01, p01.w, w11 * p11.w))) - c3);
}

// NCHW -> NHWC transpose of lookup_feats so each bilinear tap is 64B contiguous.
__global__ __launch_bounds__(TPB)
void cv_transpose_kernel(const float* __restrict__ look, float* __restrict__ lookT) {
    const int idx = blockIdx.x * blockDim.x + threadIdx.x;   // over B*H*W
    if (idx >= BB * HWN) return;
    const int b = idx / HWN;
    const int pix = idx - b * HWN;
    const float* src = look + (size_t)b * CC * HWN + pix;
    float v[CC];
#pragma unroll
    for (int c = 0; c < CC; ++c) v[c] = src[(size_t)c * HWN];
    float4* dst = reinterpret_cast<float4*>(lookT + (size_t)idx * CC);
    dst[0] = make_float4(v[0],  v[1],  v[2],  v[3]);
    dst[1] = make_float4(v[4],  v[5],  v[6],  v[7]);
    dst[2] = make_float4(v[8],  v[9],  v[10], v[11]);
    dst[3] = make_float4(v[12], v[13], v[14], v[15]);
}

// Main cost-volume kernel: one thread = one pixel x DCHUNK depths.
template <bool TRANS>
__global__ __launch_bounds__(TPB)
void cv_main_kernel(const float* __restrict__ cur,
                    const float* __restrict__ look,    // NCHW (fallback path)
                    const float* __restrict__ lookT,   // NHWC (fast path)
                    const float* __restrict__ pose,    // B x 4x4
                    const float* __restrict__ Kmat,    // B x 4x4
                    const float* __restrict__ iK,      // B x 4x4
                    float* __restrict__ out)           // B x D x H x W
{
    const int tid = threadIdx.x;
    const int p0  = blockIdx.x * TPB;
    const int pix = p0 + tid;
    const int b   = blockIdx.z;                 // wave-uniform -> scalar loads below
    const int d0  = blockIdx.y * DCHUNK;

    const float* curb = cur + (size_t)b * CC * HWN;

    float cf[CC];
#if HAVE_ASYNC_LDS
    // Cooperatively stage this block's 256-pixel x 16-channel current-feature
    // tile (16 KB) into LDS via async global->LDS b128 copies (ASYNCcnt).
    __shared__ float s_cur[CC * TPB];
#pragma unroll
    for (int i = 0; i < 4; ++i) {
        const int flat  = i * TPB + tid;        // 0..1023, each moves 16B
        const int c     = flat >> 6;            // channel 0..15
        const int chunk = flat & 63;            // 16B chunk within channel row
        int sp = p0 + chunk * 4;
        if (sp > HWN - 4) sp = HWN - 4;         // clamp: only affects invalid pixels
        async_copy_b128(curb + (size_t)c * HWN + sp, &s_cur[c * TPB + chunk * 4]);
    }
    wait_async_zero();
    __syncthreads();
    if (pix >= HWN) return;
#pragma unroll
    for (int c = 0; c < CC; ++c) cf[c] = s_cur[c * TPB + tid];   // ds_load, conflict-free
#else
    if (pix >= HWN) return;
#pragma unroll
    for (int c = 0; c < CC; ++c) cf[c] = curb[(size_t)c * HWN + pix];
#endif

    // ---- projection algebra (tiny, once per thread; b uniform per block) ----
    // P = (K @ T)[:3,:4];  A = P[:,:3] @ invK[:3,:3];  t = P[:,3]
    const float* Kb  = Kmat + b * 16;
    const float* Tb  = pose + b * 16;
    const float* iKb = iK   + b * 16;
    float P[3][4];
#pragma unroll
    for (int i = 0; i < 3; ++i)
#pragma unroll
        for (int j = 0; j < 4; ++j) {
            float s = 0.f;
#pragma unroll
            for (int k = 0; k < 4; ++k) s = fmaf(Kb[i * 4 + k], Tb[k * 4 + j], s);
            P[i][j] = s;
        }
    float A[3][3];
#pragma unroll
    for (int i = 0; i < 3; ++i)
#pragma unroll
        for (int j = 0; j < 3; ++j) {
            float s = 0.f;
#pragma unroll
            for (int k = 0; k < 3; ++k) s = fmaf(P[i][k], iKb[k * 4 + j], s);
            A[i][j] = s;
        }
    const float tx = P[0][3], ty = P[1][3], tz = P[2][3];

    const float fx = (float)(pix % WW);
    const float fy = (float)(pix / WW);
    const float bx = fmaf(A[0][0], fx, fmaf(A[0][1], fy, A[0][2]));
    const float by = fmaf(A[1][0], fx, fmaf(A[1][1], fy, A[1][2]));
    const float bz = fmaf(A[2][0], fx, fmaf(A[2][1], fy, A[2][2]));

    const float* lookb  = look  + (size_t)b * CC * HWN;
    const float* lookTb = TRANS ? (lookT + (size_t)b * HWN * CC) : nullptr;
    if (TRANS) {
        __builtin_prefetch(lookTb + (size_t)pix * CC, 0, 3);     // global_prefetch_b8
    }

    float* outb = out + (size_t)b * DD * HWN + pix;

    for (int dd = 0; dd < DCHUNK; ++dd) {
        const int   d     = d0 + dd;
        const float depth = fmaf(DSTEP, (float)d, MIN_DEPTH);
        const float pz    = fmaf(depth, bz, tz);
        const float inv   = 1.0f / (pz + EPS);
        const float gx    = fmaf(depth, bx, tx) * inv;
        const float gy    = fmaf(depth, by, ty) * inv;

        const float edge = (gx >= 2.0f && gx <= (float)(WW - 2) &&
                            gy >= 2.0f && gy <= (float)(HH - 2)) ? 1.0f : 0.0f;

        const float x0f = floorf(gx), y0f = floorf(gy);
        const float wx1 = gx - x0f,   wy1 = gy - y0f;
        const float wx0 = 1.0f - wx1, wy0 = 1.0f - wy1;

        const bool vx0 = (x0f >= 0.0f)        && (x0f <= (float)(WW - 1));
        const bool vx1 = (x0f + 1.0f >= 0.0f) && (x0f + 1.0f <= (float)(WW - 1));
        const bool vy0 = (y0f >= 0.0f)        && (y0f <= (float)(HH - 1));
        const bool vy1 = (y0f + 1.0f >= 0.0f) && (y0f + 1.0f <= (float)(HH - 1));

        const float w00 = wx0 * wy0 * ((vx0 && vy0) ? 1.0f : 0.0f);
        const float w10 = wx1 * wy0 * ((vx1 && vy0) ? 1.0f : 0.0f);
        const float w01 = wx0 * wy1 * ((vx0 && vy1) ? 1.0f : 0.0f);
        const float w11 = wx1 * wy1 * ((vx1 && vy1) ? 1.0f : 0.0f);

        const int x0 = iclamp((int)x0f, 0, WW - 1);
        const int x1 = iclamp((int)x0f + 1, 0, WW - 1);
        const int y0 = iclamp((int)y0f, 0, HH - 1);
        const int y1 = iclamp((int)y0f + 1, 0, HH - 1);

        float sum = 0.0f;
        if (TRANS) {
            // 4 taps x 16 channels, each tap = 4x global_load_b128 (64B/lane)
            const float4* t00 = reinterpret_cast<const float4*>(lookTb + (size_t)(y0 * WW + x0) * CC);
            const float4* t10 = reinterpret_cast<const float4*>(lookTb + (size_t)(y0 * WW + x1) * CC);
            const float4* t01 = reinterpret_cast<const float4*>(lookTb + (size_t)(y1 * WW + x0) * CC);
            const float4* t11 = reinterpret_cast<const float4*>(lookTb + (size_t)(y1 * WW + x1) * CC);
            acc_tap4(t00[0], t10[0], t01[0], t11[0], cf[0],  cf[1],  cf[2],  cf[3],  w00, w10, w01, w11, sum);
            acc_tap4(t00[1], t10[1], t01[1], t11[1], cf[4],  cf[5],  cf[6],  cf[7],  w00, w10, w01, w11, sum);
            acc_tap4(t00[2], t10[2], t01[2], t11[2], cf[8],  cf[9],  cf[10], cf[11], w00, w10, w01, w11, sum);
            acc_tap4(t00[3], t10[3], t01[3], t11[3], cf[12], cf[13], cf[14], cf[15], w00, w10, w01, w11, sum);
        } else {
            const int o00 = y0 * WW + x0, o10 = y0 * WW + x1;
            const int o01 = y1 * WW + x0, o11 = y1 * WW + x1;
#pragma unroll
            for (int c = 0; c < CC; ++c) {
                const float* lc = lookb + (size_t)c * HWN;
                const float wv = fmaf(w00, lc[o00], fmaf(w10, lc[o10],
                                 fmaf(w01, lc[o01], w11 * lc[o11])));
                sum += fabsf(wv - cf[c]);
            }
        }
        outb[(size_t)d * HWN] = sum * (edge * (1.0f / (float)CC));
    }
}

// Second pass: per-pixel max over depth; zeros -> max (the "missing" fixup).
__global__ __launch_bounds__(TPB)
void cv_fix_kernel(float* __restrict__ out) {
    const int idx = blockIdx.x * blockDim.x + threadIdx.x;   // over B*H*W
    if (idx >= BB * HWN) return;
    const int b   = idx / HWN;
    const int pix = idx - b * HWN;
    float* p = out + (size_t)b * DD * HWN + pix;
    float m = 0.0f;                       // diffs are >= 0
    for (int d = 0; d < DD; ++d) m = fmaxf(m, p[(size_t)d * HWN]);
    for (int d = 0; d < DD; ++d) {
        const float v = p[(size_t)d * HWN];
        if (v == 0.0f) p[(size_t)d * HWN] = m;
    }
}

extern "C" void kernel_launch(void* const* d_in, const int* in_sizes, int n_in,
                              void* d_out, int out_size, void* d_ws, size_t ws_size,
                              hipStream_t stream) {
    const float* cur  = (const float*)d_in[0];   // current_feats  (B,C,H,W)
    const float* look = (const float*)d_in[1];   // lookup_feats   (B,C,H,W)
    const float* pose = (const float*)d_in[2];   // rel_pose       (B,4,4)
    const float* Kmat = (const float*)d_in[3];   // K              (B,4,4)
    const float* iK   = (const float*)d_in[4];   // invK           (B,4,4)
    float* out = (float*)d_out;                  // (B,D,H,W)

    const size_t needT = (size_t)BB * HWN * CC * sizeof(float);  // ~2.1 MB
    const bool trans = (d_ws != nullptr) && (ws_size >= needT);
    float* lookT = (float*)d_ws;

    const dim3 blk(TPB);
    const dim3 grid_main((HWN + TPB - 1) / TPB, DD / DCHUNK, BB);
    const dim3 grid_flat((BB * HWN + TPB - 1) / TPB);

    if (trans) {
        cv_transpose_kernel<<<grid_flat, blk, 0, stream>>>(look, lookT);
        cv_main_kernel<true><<<grid_main, blk, 0, stream>>>(cur, look, lookT,
                                                            pose, Kmat, iK, out);
    } else {
        cv_main_kernel<false><<<grid_main, blk, 0, stream>>>(cur, look, look,
                                                             pose, Kmat, iK, out);
    }
    cv_fix_kernel<<<grid_flat, blk, 0, stream>>>(out);
}